// Model_91302414778568
// MI455X (gfx1250) — compile-verified
//
#include <hip/hip_runtime.h>
#include <hip/hip_bf16.h>
#include <math.h>

// ---------------------------------------------------------------------------
// Problem constants (from reference)
// ---------------------------------------------------------------------------
#define kB      2
#define kN      4096
#define kNS     12
#define kF      14        // 2 + NS features in d1
#define kK      8         // knn
#define kNG     6
#define kHEADS  2
#define kNBLOCK 3
#define kSLOPE  0.2f
#define kBN     (kB * kN)

typedef float v2f __attribute__((ext_vector_type(2)));
typedef float v4f __attribute__((ext_vector_type(4)));
typedef float v8f __attribute__((ext_vector_type(8)));

__device__ __forceinline__ float lrelu(float v) { return v > 0.f ? v : kSLOPE * v; }

// Intra-wave LDS ordering: LDS ops of a wave complete in order; a single
// s_wait_dscnt 0 (plus compiler fence) makes cross-lane stores visible to
// subsequent loads of the same wave. No workgroup barrier needed.
__device__ __forceinline__ void wave_lds_fence() {
    asm volatile("s_wait_dscnt 0x0" ::: "memory");
    __builtin_amdgcn_wave_barrier();
}

// ---------------------------------------------------------------------------
// Kernel 1: pack features into WMMA-lane-permuted, 16-padded layout and
// compute row squared norms in one pass.
//   xp[n][h*8 + 2c + j] = x[n][4c + 2h + j]   (h=0/1, c=0..3, j=0/1; f>=14 -> 0)
// With this layout a lane's whole A/B fragment is two aligned b128 loads.
// ---------------------------------------------------------------------------
__global__ void pack_kernel(const float* __restrict__ x,
                            float* __restrict__ xp,
                            float* __restrict__ d2) {
    int tid = blockIdx.x * blockDim.x + threadIdx.x;
    if (tid >= kBN) return;
    const float* xr = x + (size_t)tid * kF;
    float buf[kF];
    float s = 0.f;
#pragma unroll
    for (int f = 0; f < kF; ++f) { buf[f] = xr[f]; s += buf[f] * buf[f]; }
    float* o = xp + (size_t)tid * 16;
#pragma unroll
    for (int h = 0; h < 2; ++h)
#pragma unroll
        for (int c = 0; c < 4; ++c)
#pragma unroll
            for (int j = 0; j < 2; ++j) {
                const int f = 4 * c + 2 * h + j;
                o[h * 8 + 2 * c + j] = (f < kF) ? buf[f] : 0.f;
            }
    d2[tid] = s;
}

// ---------------------------------------------------------------------------
// Kernel 2: fused KNN via WMMA Gram tiles + on-chip top-8.
// One wave32 per 16-row strip, 4x V_WMMA_F32_16X16X4_F32 per 16x16 tile.
// Writer lanes fold the column norm and self-exclusion into the LDS spill;
// scan lanes preload all 8 candidates (latency overlapped under one DS wait)
// and keep a running (max, slot) so most candidates cost 1 add + 1 cmp.
// ---------------------------------------------------------------------------
__global__ __launch_bounds__(256) void knn_kernel(const float* __restrict__ xp,
                                                  const float* __restrict__ d2,
                                                  int* __restrict__ idx) {
    __shared__ float lds_tile[8][16][17];   // per-wave (d2col - 2*gram) tile
    __shared__ float lds_md[8][16][16];     // merge: candidate distances
    __shared__ int   lds_mi[8][16][16];     // merge: candidate indices

    const int wave  = threadIdx.x >> 5;
    const int lane  = threadIdx.x & 31;
    const int strip = blockIdx.x * 8 + wave;          // 0 .. B*(N/16)-1
    const int b     = strip / (kN / 16);
    const int m0    = (strip % (kN / 16)) * 16;

    const float* xb  = xp + (size_t)b * kN * 16;
    const float* d2b = d2 + (size_t)b * kN;

    const int hsel = (lane < 16) ? 0 : 1;   // which K-half this lane holds

    // ---- load A fragment: two b128 loads thanks to the packed layout ----
    v2f a[4];
    {
        const float* rp = xb + (size_t)(m0 + (lane & 15)) * 16 + hsel * 8;
        v4f a0 = *(const v4f*)rp;
        v4f a1 = *(const v4f*)(rp + 4);
        a[0] = v2f{a0.x, a0.y}; a[1] = v2f{a0.z, a0.w};
        a[2] = v2f{a1.x, a1.y}; a[3] = v2f{a1.z, a1.w};
    }

    // writer-lane constants (spill side)
    const int wcol  = lane & 15;
    const int rbase = (lane < 16) ? 0 : 8;
    // scanner-lane constants: 2 lanes per row, 8 cols each
    const int   myrow = lane >> 1;
    const int   c0    = (lane & 1) * 8;
    const float d2row = d2b[m0 + myrow];

    float bestd[kK];
    int   besti[kK];
#pragma unroll
    for (int t = 0; t < kK; ++t) { bestd[t] = 3.4e38f; besti[t] = 0; }
    float curmax  = 3.4e38f;
    int   curslot = 0;

    for (int jt = 0; jt < kN / 16; ++jt) {
        const int cb = jt * 16;

        // prefetch next column tile (emits global_prefetch_b8)
        if (jt + 1 < kN / 16)
            __builtin_prefetch(&xb[(size_t)(cb + 16 + wcol) * 16], 0, 3);

        // ---- load B fragment: two aligned b128 loads ----
        v2f bt[4];
        {
            const float* rp = xb + (size_t)(cb + wcol) * 16 + hsel * 8;
            v4f b0 = *(const v4f*)rp;
            v4f b1 = *(const v4f*)(rp + 4);
            bt[0] = v2f{b0.x, b0.y}; bt[1] = v2f{b0.z, b0.w};
            bt[2] = v2f{b1.x, b1.y}; bt[3] = v2f{b1.z, b1.w};
        }

        // ---- Gram tile: 4 chained f32 WMMAs (K = 16, chunks of 4) ----
        v8f acc = {};
#pragma unroll
        for (int c = 0; c < 4; ++c) {
            acc = __builtin_amdgcn_wmma_f32_16x16x4_f32(
                /*neg_a=*/false, a[c], /*neg_b=*/false, bt[c],
                /*c_mod=*/(short)0, acc, /*reuse_a=*/false, /*reuse_b=*/false);
        }

        // ---- spill (d2col - 2g), folding self-exclusion, to wave's LDS ----
        {
            const float d2col = d2b[cb + wcol];
            const int   colg  = cb + wcol;
#pragma unroll
            for (int v = 0; v < 8; ++v) {
                float t = d2col - 2.f * acc[v];
                if (colg == m0 + rbase + v) t = 3.0e38f;   // diag (+1e9)
                lds_tile[wave][rbase + v][wcol] = t;
            }
        }
        wave_lds_fence();

        // ---- preload 8 candidates (loads overlap under one DS wait) ----
        float cand[8];
#pragma unroll
        for (int c = 0; c < 8; ++c)
            cand[c] = lds_tile[wave][myrow][c0 + c];

        // ---- top-8 scan: dist = d2row + cand; rare-path insert ----
#pragma unroll
        for (int c = 0; c < 8; ++c) {
            const float dist = d2row + cand[c];
            if (dist < curmax) {
                bestd[curslot] = dist;
                besti[curslot] = cb + c0 + c;
                curmax  = bestd[0];
                curslot = 0;
#pragma unroll
                for (int t = 1; t < kK; ++t)
                    if (bestd[t] > curmax) { curmax = bestd[t]; curslot = t; }
            }
        }
        wave_lds_fence();   // reads done before next iteration's spill
    }

    // ---- merge the two half-row candidate sets, emit 8 smallest ----
#pragma unroll
    for (int t = 0; t < kK; ++t) {
        lds_md[wave][myrow][(lane & 1) * 8 + t] = bestd[t];
        lds_mi[wave][myrow][(lane & 1) * 8 + t] = besti[t];
    }
    wave_lds_fence();
    if ((lane & 1) == 0) {
        float dd[16]; int ii[16];
#pragma unroll
        for (int t = 0; t < 16; ++t) {
            dd[t] = lds_md[wave][myrow][t];
            ii[t] = lds_mi[wave][myrow][t];
        }
        int* out = idx + ((size_t)b * kN + (m0 + myrow)) * kK;
#pragma unroll
        for (int s = 0; s < kK; ++s) {
            int mn = 0; float mnv = dd[0];
#pragma unroll
            for (int t = 1; t < 16; ++t)
                if (dd[t] < mnv) { mnv = dd[t]; mn = t; }
            out[s] = ii[mn];
            dd[mn] = 3.4e38f;
        }
    }
}

// ---------------------------------------------------------------------------
// GAT conv phase A: h[n,h,d] = sum_f x[n,f] * W[f,h,d] ; attention logits
// es[n,h] = h . a_src, ed[n,h] = h . a_dst.
// Explicit v4f on W rows / h rows (all 16B aligned for D in {12,32}).
// ---------------------------------------------------------------------------
template <int IN, int D>
__global__ void gat_phase_a(const float* __restrict__ x, int x_stride, int x_off,
                            const float* __restrict__ W,
                            const float* __restrict__ a_s,
                            const float* __restrict__ a_d,
                            float* __restrict__ h,
                            float* __restrict__ es, float* __restrict__ ed) {
    constexpr int Q = D / 4;
    int tid = blockIdx.x * blockDim.x + threadIdx.x;
    if (tid >= kBN) return;
    const float* xr = x + (size_t)tid * x_stride + x_off;
    float xin[IN];
#pragma unroll
    for (int f = 0; f < IN; ++f) xin[f] = xr[f];
#pragma unroll
    for (int hh = 0; hh < kHEADS; ++hh) {
        v4f hacc[Q];
#pragma unroll
        for (int q = 0; q < Q; ++q) hacc[q] = v4f{0.f, 0.f, 0.f, 0.f};
#pragma unroll
        for (int f = 0; f < IN; ++f) {
            const float xv = xin[f];
            const v4f* wr = (const v4f*)(W + ((size_t)f * kHEADS + hh) * D);
#pragma unroll
            for (int q = 0; q < Q; ++q) hacc[q] += xv * wr[q];
        }
        const v4f* asr = (const v4f*)(a_s + hh * D);
        const v4f* adr = (const v4f*)(a_d + hh * D);
        v4f se = v4f{0.f, 0.f, 0.f, 0.f};
        v4f de = v4f{0.f, 0.f, 0.f, 0.f};
        v4f* hr = (v4f*)(h + ((size_t)tid * kHEADS + hh) * D);
#pragma unroll
        for (int q = 0; q < Q; ++q) {
            hr[q] = hacc[q];
            se += hacc[q] * asr[q];
            de += hacc[q] * adr[q];
        }
        es[tid * kHEADS + hh] = se.x + se.y + se.z + se.w;
        ed[tid * kHEADS + hh] = de.x + de.y + de.z + de.w;
    }
}

// ---------------------------------------------------------------------------
// GAT conv phase B: gather 8 neighbors, lrelu + softmax over k, combine heads.
// Neighbor rows gathered as aligned v4f (global_load_b128).
// ---------------------------------------------------------------------------
template <int D>
__global__ void gat_phase_b(const float* __restrict__ h,
                            const float* __restrict__ es,
                            const float* __restrict__ ed,
                            const int* __restrict__ idx,
                            float* __restrict__ out, int leaky_out) {
    constexpr int Q = D / 4;
    int tid = blockIdx.x * blockDim.x + threadIdx.x;
    if (tid >= kBN) return;
    const int  b  = tid / kN;
    const int* nb = idx + (size_t)tid * kK;

    int jn[kK];
#pragma unroll
    for (int k = 0; k < kK; ++k) jn[k] = b * kN + nb[k];

    float alpha[kK][kHEADS];
#pragma unroll
    for (int hh = 0; hh < kHEADS; ++hh) {
        const float esrc = es[tid * kHEADS + hh];
        float e[kK];
        float mx = -3.4e38f;
#pragma unroll
        for (int k = 0; k < kK; ++k) {
            float v = lrelu(esrc + ed[jn[k] * kHEADS + hh]);
            e[k] = v;
            mx = fmaxf(mx, v);
        }
        float s = 0.f;
#pragma unroll
        for (int k = 0; k < kK; ++k) { e[k] = expf(e[k] - mx); s += e[k]; }
        const float inv = 1.f / s;
#pragma unroll
        for (int k = 0; k < kK; ++k) alpha[k][hh] = e[k] * inv;
    }

    v4f acc[Q];
#pragma unroll
    for (int q = 0; q < Q; ++q) acc[q] = v4f{0.f, 0.f, 0.f, 0.f};
#pragma unroll
    for (int k = 0; k < kK; ++k) {
#pragma unroll
        for (int hh = 0; hh < kHEADS; ++hh) {
            const float w = alpha[k][hh];
            const v4f* hr = (const v4f*)(h + ((size_t)jn[k] * kHEADS + hh) * D);
#pragma unroll
            for (int q = 0; q < Q; ++q) acc[q] += w * hr[q];
        }
    }
    const float sc = 1.f / kHEADS;
    v4f* o = (v4f*)(out + (size_t)tid * D);
#pragma unroll
    for (int q = 0; q < Q; ++q) {
        v4f v = acc[q] * sc;
        if (leaky_out) {
            v.x = lrelu(v.x); v.y = lrelu(v.y);
            v.z = lrelu(v.z); v.w = lrelu(v.w);
        }
        o[q] = v;
    }
}

// ---------------------------------------------------------------------------
// Gumbel hard select: forward value of (hard + y - stopgrad(y)) == hard, so
// selection reduces to argmax(logits + gumbel). Writes x1/x2 + output slice.
// ---------------------------------------------------------------------------
__global__ void gumbel_select_kernel(const float* __restrict__ w1buf,
                                     const float* __restrict__ gumbel,
                                     const float* __restrict__ d1cur,
                                     int blocki,
                                     float* __restrict__ x1,
                                     float* __restrict__ x2,
                                     float* __restrict__ out_slice) {
    int tid = blockIdx.x * blockDim.x + threadIdx.x;
    if (tid >= kBN) return;
    const float* w = w1buf + (size_t)tid * kNS;
#pragma unroll
    for (int s = 0; s < 2; ++s) {
        const float* g = gumbel + ((size_t)(blocki * 2 + s) * kBN + tid) * kNG;
        int   best = 0;
        float bv   = -3.4e38f;
#pragma unroll
        for (int gg = 0; gg < kNG; ++gg) {
            float v = w[s + 2 * gg] + g[gg];
            if (v > bv) { bv = v; best = gg; }
        }
        const float xv = d1cur[(size_t)tid * kF + 2 + s + 2 * best];
        (s == 0 ? x1 : x2)[tid] = xv;
        out_slice[(size_t)tid * 2 + s] = xv;
    }
}

// ---------------------------------------------------------------------------
// Build x1f_in[j] = |d1[2+j] - x_{j&1}|  (interleaved dxa/dxb)
// ---------------------------------------------------------------------------
__global__ void build_x1f_kernel(const float* __restrict__ d1cur,
                                 const float* __restrict__ x1,
                                 const float* __restrict__ x2,
                                 float* __restrict__ x1f) {
    int tid = blockIdx.x * blockDim.x + threadIdx.x;
    if (tid >= kBN) return;
    const float a = x1[tid], b2 = x2[tid];
#pragma unroll
    for (int g = 0; g < kNG; ++g) {
        x1f[(size_t)tid * kNS + 2 * g]     = fabsf(d1cur[(size_t)tid * kF + 2 + 2 * g] - a);
        x1f[(size_t)tid * kNS + 2 * g + 1] = fabsf(d1cur[(size_t)tid * kF + 3 + 2 * g] - b2);
    }
}

// ---------------------------------------------------------------------------
// Pairwise softmax reweight: d1[2+j] = w0*ch0[j] + (1-w0)*d1[2+j],
// w0 = softmax(x_ws[j], d_ws[j])[0]; ch0[j] = (j odd ? x2 : x1).
// Columns 0,1 (xy) are untouched, matching the reference concat.
// ---------------------------------------------------------------------------
__global__ void reweight_kernel(float* __restrict__ d1cur,
                                const float* __restrict__ xws,
                                const float* __restrict__ dws,
                                const float* __restrict__ x1,
                                const float* __restrict__ x2) {
    int tid = blockIdx.x * blockDim.x + threadIdx.x;
    if (tid >= kBN) return;
    const float c1 = x1[tid], c2 = x2[tid];
#pragma unroll
    for (int j = 0; j < kNS; ++j) {
        const float xw = xws[(size_t)tid * kNS + j];
        const float dw = dws[(size_t)tid * kNS + j];
        const float m  = fmaxf(xw, dw);
        const float ex = expf(xw - m);
        const float eD = expf(dw - m);
        const float w0 = ex / (ex + eD);
        const float c0 = (j & 1) ? c2 : c1;
        const float od = d1cur[(size_t)tid * kF + 2 + j];
        d1cur[(size_t)tid * kF + 2 + j] = w0 * c0 + (1.f - w0) * od;
    }
}

// ---------------------------------------------------------------------------
// Host orchestration
// ---------------------------------------------------------------------------
namespace {

struct FeBufs {
    float *h64, *es, *ed, *hmid;
};

inline void run_fe(const float* x, int xstride, int xoff, int m,
                   const float* W1, const float* a1s, const float* a1d,
                   const float* W2, const float* a2s, const float* a2d,
                   const int* idx, const FeBufs& wsb,
                   float* out, int out_leaky, hipStream_t stream) {
    const int TPB = 256, NB = (kBN + TPB - 1) / TPB;
    const float* W1m  = W1  + (size_t)m * kNS * kHEADS * 32;
    const float* a1sm = a1s + (size_t)m * kHEADS * 32;
    const float* a1dm = a1d + (size_t)m * kHEADS * 32;
    const float* W2m  = W2  + (size_t)m * 32 * kHEADS * kNS;
    const float* a2sm = a2s + (size_t)m * kHEADS * kNS;
    const float* a2dm = a2d + (size_t)m * kHEADS * kNS;

    // conv1 (IN=12 -> D=32), h = lrelu(gat_conv1)
    gat_phase_a<kNS, 32><<<NB, TPB, 0, stream>>>(x, xstride, xoff, W1m, a1sm, a1dm,
                                                 wsb.h64, wsb.es, wsb.ed);
    gat_phase_b<32><<<NB, TPB, 0, stream>>>(wsb.h64, wsb.es, wsb.ed, idx,
                                            wsb.hmid, /*leaky=*/1);
    // conv2 (IN=32 -> D=12)
    gat_phase_a<32, kNS><<<NB, TPB, 0, stream>>>(wsb.hmid, 32, 0, W2m, a2sm, a2dm,
                                                 wsb.h64, wsb.es, wsb.ed);
    gat_phase_b<kNS><<<NB, TPB, 0, stream>>>(wsb.h64, wsb.es, wsb.ed, idx,
                                             out, out_leaky);
}

} // namespace

extern "C" void kernel_launch(void* const* d_in, const int* in_sizes, int n_in,
                              void* d_out, int out_size, void* d_ws, size_t ws_size,
                              hipStream_t stream) {
    (void)in_sizes; (void)n_in; (void)out_size; (void)ws_size;

    const float* d1_in  = (const float*)d_in[0];
    const float* gumbel = (const float*)d_in[1];
    const float* W1     = (const float*)d_in[2];
    const float* a1s    = (const float*)d_in[3];
    const float* a1d    = (const float*)d_in[4];
    const float* W2     = (const float*)d_in[5];
    const float* a2s    = (const float*)d_in[6];
    const float* a2d    = (const float*)d_in[7];
    float* out = (float*)d_out;

    // ---- workspace carve-up ----
    char* p = (char*)d_ws;
    auto carve = [&](size_t elems) { float* r = (float*)p; p += elems * sizeof(float); return r; };
    float* d1cur = carve((size_t)kBN * kF);
    float* xp    = carve((size_t)kBN * 16);   // packed/permuted features
    float* d2    = carve(kBN);
    int*   idx   = (int*)carve((size_t)kBN * kK);
    float* feats = carve((size_t)kBN * kNS);
    float* w1b   = carve((size_t)kBN * kNS);
    float* xf    = carve((size_t)kBN * kNS);
    float* xws   = carve((size_t)kBN * kNS);
    float* dwsb  = carve((size_t)kBN * kNS);
    float* x1f   = carve((size_t)kBN * kNS);
    float* x1    = carve(kBN);
    float* x2    = carve(kBN);
    FeBufs fb;
    fb.h64  = carve((size_t)kBN * kHEADS * 32);
    fb.es   = carve((size_t)kBN * kHEADS);
    fb.ed   = carve((size_t)kBN * kHEADS);
    fb.hmid = carve((size_t)kBN * 32);

    hipMemcpyAsync(d1cur, d1_in, (size_t)kBN * kF * sizeof(float),
                   hipMemcpyDeviceToDevice, stream);

    const int TPB = 256, NB = (kBN + TPB - 1) / TPB;
    const int KNN_BLOCKS = (kB * (kN / 16)) / 8;   // 8 waves/block, 1 strip/wave

    for (int i = 0; i < kNBLOCK; ++i) {
        // pack + norms, then KNN (WMMA Gram + fused top-8)
        pack_kernel<<<NB, TPB, 0, stream>>>(d1cur, xp, d2);
        knn_kernel<<<KNN_BLOCKS, 256, 0, stream>>>(xp, d2, idx);

        // feats = lrelu(_fe(d1[:,:,2:], idx, 2i))
        run_fe(d1cur, kF, 2, 2 * i, W1, a1s, a1d, W2, a2s, a2d,
               idx, fb, feats, /*leaky=*/1, stream);
        // w1 = _fe(feats, idx, 2i+1)
        run_fe(feats, kNS, 0, 2 * i + 1, W1, a1s, a1d, W2, a2s, a2d,
               idx, fb, w1b, /*leaky=*/0, stream);

        // gumbel hard-select -> x1, x2, output slice i
        gumbel_select_kernel<<<NB, TPB, 0, stream>>>(
            w1b, gumbel, d1cur, i, x1, x2, out + (size_t)i * kBN * 2);

        if (i == kNBLOCK - 1) break;

        // x1f_in = interleaved |d1 - x|
        build_x1f_kernel<<<NB, TPB, 0, stream>>>(d1cur, x1, x2, x1f);
        // xf = lrelu(_fe(x1f_in, idx, 6+3i))
        run_fe(x1f, kNS, 0, 6 + 3 * i, W1, a1s, a1d, W2, a2s, a2d,
               idx, fb, xf, /*leaky=*/1, stream);
        // x_ws = _fe(xf, idx, 6+3i+1)
        run_fe(xf, kNS, 0, 6 + 3 * i + 1, W1, a1s, a1d, W2, a2s, a2d,
               idx, fb, xws, /*leaky=*/0, stream);
        // d_ws = _fe(feats, idx, 6+3i+2)
        run_fe(feats, kNS, 0, 6 + 3 * i + 2, W1, a1s, a1d, W2, a2s, a2d,
               idx, fb, dwsb, /*leaky=*/0, stream);

        // d1 <- concat(xy, softmax-blend(ch0, d1[2:]))
        reweight_kernel<<<NB, TPB, 0, stream>>>(d1cur, xws, dwsb, x1, x2);
    }
}